// GNNLayer_86242943303880
// MI455X (gfx1250) — compile-verified
//
#include <hip/hip_runtime.h>

#define NN 8192
#define DD 256

typedef __attribute__((ext_vector_type(2))) float v2f;
typedef __attribute__((ext_vector_type(4))) float f4;
typedef __attribute__((ext_vector_type(8))) float v8f;

// ---------------------------------------------------------------------------
// CDNA5 fp32 WMMA: D(16x16) = A(16x4) * B(4x16) + C   (wave32, VOP3P)
// ---------------------------------------------------------------------------
__device__ __forceinline__ v8f wmma4(v2f a, v2f b, v8f c) {
  return __builtin_amdgcn_wmma_f32_16x16x4_f32(false, a, false, b, (short)0, c,
                                               false, false);
}

// A-fragment from a row-major [M x K] tile (also B-fragment of a transposed
// operand, B(k,n) = Mrow-major[n][k]).  ld must be even, k0 a multiple of 4,
// so the two floats are one aligned ds_load_b64.
// Layout (ISA 7.12.2): lane L holds M = r0 + L%16, K = k0 + 2*(L/16) + {0,1}.
__device__ __forceinline__ v2f fragA(const float* p, int r0, int k0, int ld,
                                     int lane) {
  return *(const v2f*)(p + (r0 + (lane & 15)) * ld + k0 + ((lane >> 4) << 1));
}

// B-fragment from a row-major [K x N] tile (also A-fragment of a transposed
// operand: A(m,k) = Krow-major[k][m]).
// Layout: lane L holds N = n0 + L%16, K = k0 + 2*(L/16) + {0,1}.
__device__ __forceinline__ v2f fragB(const float* p, int k0, int n0, int ld,
                                     int lane) {
  const float* q = p + (k0 + ((lane >> 4) << 1)) * ld + n0 + (lane & 15);
  v2f b;
  b.x = q[0];
  b.y = q[ld];
  return b;
}

__device__ __forceinline__ void lds_st4(float* dst, f4 v) {
  *(f4*)dst = v;  // dst 16B aligned by construction
}

// ---------------------------------------------------------------------------
// hb1[i] = head[i] . b1 ; db2[i] = dep[i] . b2  (one wave per row)
// ---------------------------------------------------------------------------
__global__ void __launch_bounds__(256) k_bias(const float* __restrict__ X,
                                              const float* __restrict__ Y,
                                              const float* __restrict__ b1,
                                              const float* __restrict__ b2,
                                              float* __restrict__ xb,
                                              float* __restrict__ yb) {
  int lane = threadIdx.x & 31, wave = threadIdx.x >> 5;
  int row = blockIdx.x * 8 + wave;
  const float* xr = X + (size_t)row * DD;
  const float* yr = Y + (size_t)row * DD;
  float sx = 0.f, sy = 0.f;
  for (int k = lane; k < DD; k += 32) {
    sx += xr[k] * b1[k];
    sy += yr[k] * b2[k];
  }
  for (int off = 16; off; off >>= 1) {
    sx += __shfl_down(sx, off, 32);
    sy += __shfl_down(sy, off, 32);
  }
  if (lane == 0) {
    xb[row] = sx;
    yb[row] = sy;
  }
}

// ---------------------------------------------------------------------------
// out[NN x DD] = act(X1 @ W1 [+ X2 @ W2]), W row-major [DD x DD]
// Block: 64 rows x 256 cols, 8 waves each own a 32x64 sub-panel (8 tiles).
// DUAL/LEAKY are compile-time so the hot loop is branch-free.
// ---------------------------------------------------------------------------
template <bool DUAL, bool LEAKY>
__global__ void __launch_bounds__(256)
k_gemm_dual(const float* __restrict__ X1, const float* __restrict__ W1,
            const float* __restrict__ X2, const float* __restrict__ W2,
            float* __restrict__ out) {
  constexpr int MT = 64, KT = 32;
  constexpr int XS = KT + 4;   // 36: aligned b64 A-frags, conflict-free
  constexpr int WS = DD + 8;   // 264: conflict-free B-frags (residue 8)
  __shared__ float sX1[MT * XS];
  __shared__ float sX2[DUAL ? MT * XS : 4];
  __shared__ float sW1[KT * WS];
  __shared__ float sW2[DUAL ? KT * WS : 4];

  const int tid = threadIdx.x, lane = tid & 31, wave = tid >> 5;
  const int m_base = blockIdx.x * MT;
  const int wm = (wave >> 2) * 32;  // wave row offset within panel
  const int wn = (wave & 3) * 64;   // wave col offset

  v8f acc[2][4];
  v8f zero = {0.f, 0.f, 0.f, 0.f, 0.f, 0.f, 0.f, 0.f};
  for (int i = 0; i < 2; ++i)
    for (int j = 0; j < 4; ++j) acc[i][j] = zero;

  for (int kc = 0; kc < DD; kc += KT) {
    __syncthreads();
    for (int idx = tid; idx < (MT * KT) / 4; idx += 256) {
      int row = idx >> 3, col = (idx & 7) << 2;
      lds_st4(&sX1[row * XS + col],
              *(const f4*)(X1 + (size_t)(m_base + row) * DD + kc + col));
      if (DUAL)
        lds_st4(&sX2[row * XS + col],
                *(const f4*)(X2 + (size_t)(m_base + row) * DD + kc + col));
    }
    for (int idx = tid; idx < (KT * DD) / 4; idx += 256) {
      int row = idx >> 6, col = (idx & 63) << 2;
      lds_st4(&sW1[row * WS + col],
              *(const f4*)(W1 + (size_t)(kc + row) * DD + col));
      if (DUAL)
        lds_st4(&sW2[row * WS + col],
                *(const f4*)(W2 + (size_t)(kc + row) * DD + col));
    }
    __syncthreads();

#pragma unroll
    for (int k = 0; k < KT; k += 4) {
      v2f a10 = fragA(sX1, wm, k, XS, lane);
      v2f a11 = fragA(sX1, wm + 16, k, XS, lane);
      v2f b1v[4];
#pragma unroll
      for (int nt = 0; nt < 4; ++nt)
        b1v[nt] = fragB(sW1, k, wn + nt * 16, WS, lane);
      v2f a20, a21, b2v[4];
      if (DUAL) {
        a20 = fragA(sX2, wm, k, XS, lane);
        a21 = fragA(sX2, wm + 16, k, XS, lane);
#pragma unroll
        for (int nt = 0; nt < 4; ++nt)
          b2v[nt] = fragB(sW2, k, wn + nt * 16, WS, lane);
      }
#pragma unroll
      for (int nt = 0; nt < 4; ++nt) {
        acc[0][nt] = wmma4(a10, b1v[nt], acc[0][nt]);
        acc[1][nt] = wmma4(a11, b1v[nt], acc[1][nt]);
        if (DUAL) {
          acc[0][nt] = wmma4(a20, b2v[nt], acc[0][nt]);
          acc[1][nt] = wmma4(a21, b2v[nt], acc[1][nt]);
        }
      }
    }
  }

  const int half = lane >> 4, nn = lane & 15;
  for (int mt = 0; mt < 2; ++mt)
    for (int nt = 0; nt < 4; ++nt)
      for (int r = 0; r < 8; ++r) {
        int m = m_base + wm + mt * 16 + r + 8 * half;
        int n = wn + nt * 16 + nn;
        float v = acc[mt][nt][r];
        if (LEAKY) v = (v >= 0.f) ? v : 0.1f * v;
        out[(size_t)m * DD + n] = v;
      }
}

// ---------------------------------------------------------------------------
// S[i,j] = hA[i] . dep[j] + hb1[i] + db2[j]    (raw logits, pre-softmax)
// Block owns 64 rows; sweeps columns in 64-wide chunks.  Both operands are
// row-major along K, so B-fragments of dep^T are contiguous b64 loads.
// ---------------------------------------------------------------------------
__global__ void __launch_bounds__(256)
k_logits(const float* __restrict__ hA, const float* __restrict__ dep,
         const float* __restrict__ hb1, const float* __restrict__ db2,
         float* __restrict__ S) {
  constexpr int MT = 64, NT = 64;
  constexpr int LS = DD + 4;  // 260: aligned b64 frags, conflict-free
  __shared__ float sA[MT * LS];
  __shared__ float sB[NT * LS];
  __shared__ float sb1[MT];

  const int tid = threadIdx.x, lane = tid & 31, wave = tid >> 5;
  const int i0 = blockIdx.x * MT;
  const int mt0 = (wave >> 2) * 32;  // two 16-row tiles per wave
  const int nc = (wave & 3) * 16;    // one 16-col tile per wave

  for (int idx = tid; idx < (MT * DD) / 4; idx += 256) {
    int row = idx >> 6, col = (idx & 63) << 2;
    lds_st4(&sA[row * LS + col],
            *(const f4*)(hA + (size_t)(i0 + row) * DD + col));
  }
  if (tid < MT) sb1[tid] = hb1[i0 + tid];

  for (int c = 0; c < NN / NT; ++c) {
    const int n0 = c * NT;
    __syncthreads();
    for (int idx = tid; idx < (NT * DD) / 4; idx += 256) {
      int row = idx >> 6, col = (idx & 63) << 2;
      lds_st4(&sB[row * LS + col],
              *(const f4*)(dep + (size_t)(n0 + row) * DD + col));
    }
    __syncthreads();

    v8f zero = {0.f, 0.f, 0.f, 0.f, 0.f, 0.f, 0.f, 0.f};
    v8f acc0 = zero, acc1 = zero;
#pragma unroll 8
    for (int k = 0; k < DD; k += 4) {
      v2f a0 = fragA(sA, mt0, k, LS, lane);
      v2f a1 = fragA(sA, mt0 + 16, k, LS, lane);
      v2f b = fragA(sB, nc, k, LS, lane);  // B(k,n) = dep[n][k]
      acc0 = wmma4(a0, b, acc0);
      acc1 = wmma4(a1, b, acc1);
    }

    const int half = lane >> 4, nn = lane & 15;
    const float dbv = db2[n0 + nc + nn];
    for (int r = 0; r < 8; ++r) {
      int m0 = mt0 + r + 8 * half;
      S[(size_t)(i0 + m0) * NN + n0 + nc + nn] = acc0[r] + sb1[m0] + dbv;
      S[(size_t)(i0 + m0 + 16) * NN + n0 + nc + nn] =
          acc1[r] + sb1[m0 + 16] + dbv;
    }
  }
}

// ---------------------------------------------------------------------------
// In-place row softmax over NN columns; one block per row.
// float4 row walk: 8 global_load_b128 + 8 global_store_b128 per thread.
// ---------------------------------------------------------------------------
__global__ void __launch_bounds__(256) k_softmax(float* __restrict__ S) {
  const int tid = threadIdx.x;
  f4* r4 = (f4*)(S + (size_t)blockIdx.x * NN);
  __shared__ float red[8];

  f4 v[8];
  float m = -3.402823466e38f;
#pragma unroll
  for (int i = 0; i < 8; ++i) {
    v[i] = r4[tid + (i << 8)];
    m = fmaxf(m, fmaxf(fmaxf(v[i].x, v[i].y), fmaxf(v[i].z, v[i].w)));
  }
  for (int off = 16; off; off >>= 1) m = fmaxf(m, __shfl_xor(m, off, 32));
  if ((tid & 31) == 0) red[tid >> 5] = m;
  __syncthreads();
  m = red[0];
  for (int w = 1; w < 8; ++w) m = fmaxf(m, red[w]);
  __syncthreads();

  float s = 0.f;
#pragma unroll
  for (int i = 0; i < 8; ++i) {
    v[i].x = __expf(v[i].x - m);
    v[i].y = __expf(v[i].y - m);
    v[i].z = __expf(v[i].z - m);
    v[i].w = __expf(v[i].w - m);
    s += (v[i].x + v[i].y) + (v[i].z + v[i].w);
  }
  for (int off = 16; off; off >>= 1) s += __shfl_xor(s, off, 32);
  if ((tid & 31) == 0) red[tid >> 5] = s;
  __syncthreads();
  s = 0.f;
  for (int w = 0; w < 8; ++w) s += red[w];
  const float inv = 1.f / s;
#pragma unroll
  for (int i = 0; i < 8; ++i) {
    f4 o;
    o.x = v[i].x * inv;
    o.y = v[i].y * inv;
    o.z = v[i].z * inv;
    o.w = v[i].w * inv;
    r4[tid + (i << 8)] = o;
  }
}

// ---------------------------------------------------------------------------
// head_agg = alpha^T@H + alpha@Dp - diag(alpha)*(H+Dp)
// dep_agg  = alpha@H + alpha^T@Dp - diag(alpha)*(H+Dp)
// Block owns a 32-row panel; sweeps j in 32-chunks staging both the alpha
// row-panel and alpha column-panel tiles in LDS (each alpha element is read
// twice total per layer).
// ---------------------------------------------------------------------------
__global__ void __launch_bounds__(256)
k_agg(const float* __restrict__ alpha, const float* __restrict__ H,
      const float* __restrict__ Dp, float* __restrict__ hagg,
      float* __restrict__ dagg) {
  constexpr int IT = 32, JT = 32;
  constexpr int ARS = JT + 4;  // 36: row-panel, aligned A-frags
  constexpr int ACS = JT + 8;  // 40: col-panel read K-major (fragB pattern)
  constexpr int HS = DD + 8;   // 264: conflict-free B-frags
  __shared__ float sAR[IT * ARS];  // alpha[i0+m][j0+k]
  __shared__ float sAC[JT * ACS];  // alpha[j0+k][i0+m]
  __shared__ float sH[JT * HS];
  __shared__ float sD[JT * HS];
  __shared__ float sDiag[IT];

  const int tid = threadIdx.x, lane = tid & 31, wave = tid >> 5;
  const int i0 = blockIdx.x * IT;
  const int wn = wave * 32;  // feature columns owned by this wave

  if (tid < IT) sDiag[tid] = alpha[(size_t)(i0 + tid) * NN + i0 + tid];

  v8f zero = {0.f, 0.f, 0.f, 0.f, 0.f, 0.f, 0.f, 0.f};
  v8f ah[2][2], ad[2][2];
  for (int i = 0; i < 2; ++i)
    for (int j = 0; j < 2; ++j) {
      ah[i][j] = zero;
      ad[i][j] = zero;
    }

  for (int j0 = 0; j0 < NN; j0 += JT) {
    __syncthreads();
    for (int idx = tid; idx < (IT * JT) / 4; idx += 256) {
      int row = idx >> 3, col = (idx & 7) << 2;
      lds_st4(&sAR[row * ARS + col],
              *(const f4*)(alpha + (size_t)(i0 + row) * NN + j0 + col));
      lds_st4(&sAC[row * ACS + col],
              *(const f4*)(alpha + (size_t)(j0 + row) * NN + i0 + col));
    }
    for (int idx = tid; idx < (JT * DD) / 4; idx += 256) {
      int row = idx >> 6, col = (idx & 63) << 2;
      lds_st4(&sH[row * HS + col],
              *(const f4*)(H + (size_t)(j0 + row) * DD + col));
      lds_st4(&sD[row * HS + col],
              *(const f4*)(Dp + (size_t)(j0 + row) * DD + col));
    }
    __syncthreads();

#pragma unroll
    for (int k = 0; k < JT; k += 4) {
      v2f ar0 = fragA(sAR, 0, k, ARS, lane);   // A(m,k)=alpha[i0+m][j0+k]
      v2f ar1 = fragA(sAR, 16, k, ARS, lane);
      v2f at0 = fragB(sAC, k, 0, ACS, lane);   // A(m,k)=alpha[j0+k][i0+m]
      v2f at1 = fragB(sAC, k, 16, ACS, lane);
      v2f bh[2], bd[2];
#pragma unroll
      for (int nt = 0; nt < 2; ++nt) {
        bh[nt] = fragB(sH, k, wn + nt * 16, HS, lane);
        bd[nt] = fragB(sD, k, wn + nt * 16, HS, lane);
      }
#pragma unroll
      for (int nt = 0; nt < 2; ++nt) {
        ah[0][nt] = wmma4(at0, bh[nt], ah[0][nt]);
        ah[1][nt] = wmma4(at1, bh[nt], ah[1][nt]);
        ah[0][nt] = wmma4(ar0, bd[nt], ah[0][nt]);
        ah[1][nt] = wmma4(ar1, bd[nt], ah[1][nt]);
        ad[0][nt] = wmma4(ar0, bh[nt], ad[0][nt]);
        ad[1][nt] = wmma4(ar1, bh[nt], ad[1][nt]);
        ad[0][nt] = wmma4(at0, bd[nt], ad[0][nt]);
        ad[1][nt] = wmma4(at1, bd[nt], ad[1][nt]);
      }
    }
  }

  const int half = lane >> 4, nn = lane & 15;
  for (int mt = 0; mt < 2; ++mt)
    for (int nt = 0; nt < 2; ++nt)
      for (int r = 0; r < 8; ++r) {
        int m = mt * 16 + r + 8 * half;
        int gm = i0 + m;
        int n = wn + nt * 16 + nn;
        float hd = H[(size_t)gm * DD + n] + Dp[(size_t)gm * DD + n];
        float da = sDiag[m];
        hagg[(size_t)gm * DD + n] = ah[mt][nt][r] - da * hd;
        dagg[(size_t)gm * DD + n] = ad[mt][nt][r] - da * hd;
      }
}

// ---------------------------------------------------------------------------
// Host-side layer driver
// ---------------------------------------------------------------------------
static void run_layer(const float* h, const float* d, const float* A,
                      const float* b1, const float* b2, const float* awh,
                      const float* awd, const float* cwh, const float* cwd,
                      float* alpha, float* new_h, float* new_d, float* hAbuf,
                      float* hagg, float* dagg, float* hb1, float* db2,
                      hipStream_t stream) {
  k_bias<<<NN / 8, 256, 0, stream>>>(h, d, b1, b2, hb1, db2);
  k_gemm_dual<false, false>
      <<<NN / 64, 256, 0, stream>>>(h, A, nullptr, nullptr, hAbuf);
  k_logits<<<NN / 64, 256, 0, stream>>>(hAbuf, d, hb1, db2, alpha);
  k_softmax<<<NN, 256, 0, stream>>>(alpha);
  k_agg<<<NN / 32, 256, 0, stream>>>(alpha, h, d, hagg, dagg);
  k_gemm_dual<true, true>
      <<<NN / 64, 256, 0, stream>>>(hagg, awh, h, cwh, new_h);
  k_gemm_dual<true, true>
      <<<NN / 64, 256, 0, stream>>>(dagg, awd, d, cwd, new_d);
}

extern "C" void kernel_launch(void* const* d_in, const int* in_sizes, int n_in,
                              void* d_out, int out_size, void* d_ws,
                              size_t ws_size, hipStream_t stream) {
  const float* head = (const float*)d_in[0];
  const float* dep = (const float*)d_in[1];
  const float* tp1_A = (const float*)d_in[2];
  const float* tp1_b1 = (const float*)d_in[3];
  const float* tp1_b2 = (const float*)d_in[4];
  const float* tp2_A = (const float*)d_in[5];
  const float* tp2_b1 = (const float*)d_in[6];
  const float* tp2_b2 = (const float*)d_in[7];
  const float* g1_awh = (const float*)d_in[8];
  const float* g1_awd = (const float*)d_in[9];
  const float* g1_cwh = (const float*)d_in[10];
  const float* g1_cwd = (const float*)d_in[11];
  const float* g2_awh = (const float*)d_in[12];
  const float* g2_awd = (const float*)d_in[13];
  const float* g2_cwh = (const float*)d_in[14];
  const float* g2_cwd = (const float*)d_in[15];

  // Output layout: head, dependent, alpha1, alpha2 (flat, return order)
  float* out_head = (float*)d_out;
  float* out_dep = out_head + (size_t)NN * DD;
  float* alpha1 = out_dep + (size_t)NN * DD;
  float* alpha2 = alpha1 + (size_t)NN * NN;

  // Workspace carve-up (floats): ~42 MB total
  float* ws = (float*)d_ws;
  const size_t nd = (size_t)NN * DD;
  float* hAbuf = ws;            ws += nd;
  float* hagg  = ws;            ws += nd;
  float* dagg  = ws;            ws += nd;
  float* h1    = ws;            ws += nd;
  float* d1    = ws;            ws += nd;
  float* hb1   = ws;            ws += NN;
  float* db2   = ws;            ws += NN;

  run_layer(head, dep, tp1_A, tp1_b1, tp1_b2, g1_awh, g1_awd, g1_cwh, g1_cwd,
            alpha1, h1, d1, hAbuf, hagg, dagg, hb1, db2, stream);
  run_layer(h1, d1, tp2_A, tp2_b1, tp2_b2, g2_awh, g2_awd, g2_cwh, g2_cwd,
            alpha2, out_head, out_dep, hAbuf, hagg, dagg, hb1, db2, stream);
}